// BallQuery_16733192585421
// MI455X (gfx1250) — compile-verified
//
#include <hip/hip_runtime.h>

typedef float v2f __attribute__((ext_vector_type(2)));
typedef float v8f __attribute__((ext_vector_type(8)));
typedef int   v4i __attribute__((ext_vector_type(4)));

#define NPTS   16384
#define MCTR   4096
#define CFEAT  32
#define KNN    32
#define RAD2   0.01f
#define NB     2
#define CHUNK  2048
#define NCHUNK (NPTS / CHUNK)
#define TILES  (CHUNK / 16)
#define WGT    128   // 4 waves, 16 centers per wave, 64 centers per workgroup

// ---- CDNA5 async global->LDS staging (ASYNCcnt path), guarded -------------
#if __has_builtin(__builtin_amdgcn_global_load_async_to_lds_b128)
#define HAVE_ASYNC_LDS 1
#else
#define HAVE_ASYNC_LDS 0
#endif

#if HAVE_ASYNC_LDS
__device__ __forceinline__ void async_cp16(const float* g, float* l) {
    // Signature (from clang diagnostic): (AS1 v4i* src, AS3 v4i* dst, Ii, Ii)
    auto gsrc = (__attribute__((address_space(1))) v4i*)(uintptr_t)g;
    auto lgen = (__attribute__((address_space(3))) void*)l;
    auto ldst = (__attribute__((address_space(3))) v4i*)lgen;
    __builtin_amdgcn_global_load_async_to_lds_b128(gsrc, ldst, 0, 0);
}
__device__ __forceinline__ void async_fence() {
#if __has_builtin(__builtin_amdgcn_s_wait_asynccnt)
    __builtin_amdgcn_s_wait_asynccnt(0);
#else
    asm volatile("s_wait_asynccnt 0x0" ::: "memory");
#endif
}
#else
__device__ __forceinline__ void async_fence() {}
#endif

// Stage one 2048-point chunk (x,y,z arrays) into an LDS buffer.
__device__ __forceinline__ void stage_chunk(
    const float* __restrict__ pbase, int n0,
    float* sx, float* sy, float* sz, int tid)
{
#if HAVE_ASYNC_LDS
    // 2048 floats/array = 512 float4; 128 threads -> 4 b128 per array.
#pragma unroll
    for (int s = 0; s < CHUNK / (WGT * 4); ++s) {
        const int i = (tid + s * WGT) * 4;
        async_cp16(pbase + 0 * NPTS + n0 + i, sx + i);
        async_cp16(pbase + 1 * NPTS + n0 + i, sy + i);
        async_cp16(pbase + 2 * NPTS + n0 + i, sz + i);
    }
#else
    for (int i = tid; i < CHUNK; i += WGT) {
        sx[i] = pbase[0 * NPTS + n0 + i];
        sy[i] = pbase[1 * NPTS + n0 + i];
        sz[i] = pbase[2 * NPTS + n0 + i];
    }
#endif
}

// ---------------------------------------------------------------------------
// Kernel 1: ball query via V_WMMA_F32_16X16X4_F32, double-buffered async
// LDS staging, software-pipelined B-tile loads, uniform empty-tile skip.
//   dist2 = A(16x4) x B(4x16) + C  with A row m = (-2cx,-2cy,-2cz, 1),
//   B col n = (px,py,pz,p2), C = c2 per row.
// ---------------------------------------------------------------------------
__global__ __launch_bounds__(WGT) void ballquery_wmma_kernel(
    const float* __restrict__ pts,   // [B][3][N]
    const float* __restrict__ ctr,   // [B][3][M]
    int* __restrict__ idxout)        // [B][M][K]
{
    __shared__ float sX[2][CHUNK], sY[2][CHUNK], sZ[2][CHUNK], sP2[2][CHUNK];

    const int tid  = threadIdx.x;
    const int lane = tid & 31;
    const int wave = tid >> 5;
    const int blocksPerBatch = MCTR / 64;
    const int batch = blockIdx.x / blocksPerBatch;
    const int cblk  = blockIdx.x % blocksPerBatch;
    const int row   = lane & 15;
    const bool hi   = lane >= 16;
    const int center = cblk * 64 + wave * 16 + row;

    const float* pbase = pts + batch * 3 * NPTS;
    const float* cbase = ctr + batch * 3 * MCTR;
    int* iout = idxout + (batch * MCTR + center) * KNN;

    const float cx = cbase[0 * MCTR + center];
    const float cy = cbase[1 * MCTR + center];
    const float cz = cbase[2 * MCTR + center];
    const float c2 = cx * cx + cy * cy + cz * cz;

    // A matrix (16x4 f32): VGPR0 = K0 / K2, VGPR1 = K1 / K3.
    v2f Am;
    Am.x = hi ? (-2.0f * cz) : (-2.0f * cx);
    Am.y = hi ? 1.0f         : (-2.0f * cy);

    // C matrix: C[v] on lane L = c2[ v + (L>=16 ? 8 : 0) ].
    v8f Cm;
#pragma unroll
    for (int v = 0; v < 8; ++v) {
        const int src = v + (hi ? 8 : 0);
        Cm[v] = __int_as_float(
            __builtin_amdgcn_ds_bpermute(src << 2, __float_as_int(c2)));
    }

    int  cnt = 0;
    int  firstIdx = 0;
    bool waveDone = false;

    // Prologue: stage chunk 0 into buffer 0.
    stage_chunk(pbase, 0, sX[0], sY[0], sZ[0], tid);
    async_fence();
    __syncthreads();

    for (int ci = 0; ci < NCHUNK; ++ci) {
        const int cur = ci & 1;

        // Kick off async staging of the next chunk into the other buffer;
        // its s_wait_asynccnt lands after the tile loop (fully overlapped).
        if (ci + 1 < NCHUNK)
            stage_chunk(pbase, (ci + 1) * CHUNK,
                        sX[cur ^ 1], sY[cur ^ 1], sZ[cur ^ 1], tid);

        // p^2 pass for the current buffer (LDS -> LDS).
        for (int i = tid; i < CHUNK; i += WGT) {
            const float x = sX[cur][i], y = sY[cur][i], z = sZ[cur][i];
            sP2[cur][i] = x * x + y * y + z * z;
        }
        __syncthreads();

        if (!waveDone) {   // wave-uniform: EXEC stays all-ones for WMMA
            const float* b0p = (hi ? sZ[cur]  : sX[cur]) + row;
            const float* b1p = (hi ? sP2[cur] : sY[cur]) + row;

            // Software pipeline: fetch tile t+1 while WMMA consumes tile t.
            float b0 = b0p[0];
            float b1 = b1p[0];
            for (int t = 0; t < TILES; ++t) {
                const int tn = (t + 1 < TILES) ? (t + 1) : t;
                const float nb0 = b0p[tn * 16];
                const float nb1 = b1p[tn * 16];

                v2f Bm; Bm.x = b0; Bm.y = b1;
                v8f D = __builtin_amdgcn_wmma_f32_16x16x4_f32(
                    false, Am, false, Bm, (short)0, Cm, false, false);

                unsigned mk[8];
#pragma unroll
                for (int v = 0; v < 8; ++v)
                    mk[v] = __builtin_amdgcn_ballot_w32(D[v] < RAD2);

                // Uniform (SGPR) empty-tile skip.
                const unsigned any = (mk[0] | mk[1] | mk[2] | mk[3]) |
                                     (mk[4] | mk[5] | mk[6] | mk[7]);
                if (any != 0u) {
                    // Lane L (0..15) owns row L: rows 0-7 -> mk[L] lo16,
                    // rows 8-15 -> mk[L-8] hi16.
                    const unsigned r = lane & 7;
                    const unsigned s01 = (r & 1) ? mk[1] : mk[0];
                    const unsigned s23 = (r & 1) ? mk[3] : mk[2];
                    const unsigned s45 = (r & 1) ? mk[5] : mk[4];
                    const unsigned s67 = (r & 1) ? mk[7] : mk[6];
                    const unsigned sA  = (r & 2) ? s23 : s01;
                    const unsigned sB  = (r & 2) ? s67 : s45;
                    const unsigned sS  = (r & 4) ? sB : sA;
                    unsigned rowmask = (lane < 8) ? (sS & 0xFFFFu) : (sS >> 16);

                    if (!hi && rowmask != 0u && cnt < KNN) {
                        const int nb = ci * CHUNK + t * 16;
                        do {
                            const int bpos = __builtin_ctz(rowmask);
                            const int pidx = nb + bpos;
                            if (cnt == 0) firstIdx = pidx;
                            iout[cnt] = pidx;
                            rowmask &= rowmask - 1u;
                            ++cnt;
                        } while (rowmask != 0u && cnt < KNN);
                    }
                }
                b0 = nb0; b1 = nb1;
            }
            waveDone = (__builtin_amdgcn_ballot_w32(hi || cnt >= KNN)
                        == 0xFFFFFFFFu);
        }

        async_fence();      // next chunk has landed in LDS
        __syncthreads();
    }

    // Pad remaining slots with the first hit (or 0 if none).
    if (!hi) {
        for (int k = cnt; k < KNN; ++k) iout[k] = firstIdx;
    }
}

// ---------------------------------------------------------------------------
// Kernel 2: gather + center-subtract + transposed write. m-fastest threads ->
// coalesced stores; gathers are L2-resident (pts+feat ~4.6 MB, L2 = 192 MB).
// ---------------------------------------------------------------------------
__global__ __launch_bounds__(256) void group_kernel(
    const float* __restrict__ pts,   // [B][3][N]
    const float* __restrict__ ctr,   // [B][3][M]
    const float* __restrict__ feat,  // [B][C][N]
    const int*   __restrict__ idx,   // [B][M][K]
    float* __restrict__ out)         // [B][3+C][K][M]
{
    const int g = blockIdx.x * 256 + threadIdx.x;
    const int m = g % MCTR;
    const int k = (g / MCTR) % KNN;
    const int b = g / (MCTR * KNN);

    const int n = idx[(b * MCTR + m) * KNN + k];

    const float* pb = pts  + b * 3 * NPTS;
    const float* cb = ctr  + b * 3 * MCTR;
    const float* fb = feat + b * CFEAT * NPTS;
    float* ob = out + (size_t)b * (3 + CFEAT) * KNN * MCTR
                    + (size_t)k * MCTR + m;

#pragma unroll
    for (int c = 0; c < 3; ++c)
        ob[(size_t)c * KNN * MCTR] = pb[c * NPTS + n] - cb[c * MCTR + m];
#pragma unroll 4
    for (int c = 0; c < CFEAT; ++c)
        ob[(size_t)(3 + c) * KNN * MCTR] = fb[c * NPTS + n];
}

// ---------------------------------------------------------------------------
extern "C" void kernel_launch(void* const* d_in, const int* in_sizes, int n_in,
                              void* d_out, int out_size, void* d_ws, size_t ws_size,
                              hipStream_t stream)
{
    const float* pts  = (const float*)d_in[0];  // points_coords  [B][3][N]
    const float* ctr  = (const float*)d_in[1];  // centers_coords [B][3][M]
    const float* feat = (const float*)d_in[2];  // points_features[B][C][N]
    int*   idxws = (int*)d_ws;                  // [B][M][K] = 1 MB scratch
    float* outp  = (float*)d_out;

    ballquery_wmma_kernel<<<dim3(NB * (MCTR / 64)), dim3(WGT), 0, stream>>>(
        pts, ctr, idxws);

    group_kernel<<<dim3(NB * KNN * MCTR / 256), dim3(256), 0, stream>>>(
        pts, ctr, feat, idxws, outp);
}